// BiGaBP_unfolding_iter_84731114815616
// MI455X (gfx1250) — compile-verified
//
#include <hip/hip_runtime.h>
#include <cstdint>

// Problem constants (match the reference's static dims)
#define NR_   16
#define NT_   8
#define K_    64
#define ELEMS 8192          // NR*NT*K per (b, tensor)
#define THREADS 1024        // one thread per (nr,k)

typedef int v4i __attribute__((vector_size(16)));
typedef __attribute__((address_space(1))) v4i as1_v4i;
typedef __attribute__((address_space(3))) v4i as3_v4i;

// ---- CDNA5 async global->LDS bulk copy (ASYNCcnt path) -------------------
__device__ __forceinline__ void async_g2l_b128(const float* g, float* l) {
#if __has_builtin(__builtin_amdgcn_global_load_async_to_lds_b128)
  // global_load_async_to_lds_b128: 16B/lane, tracked on ASYNCcnt,
  // bypasses the VGPR return path entirely.
  __builtin_amdgcn_global_load_async_to_lds_b128(
      (as1_v4i*)(uintptr_t)g,
      (as3_v4i*)(uint32_t)(uintptr_t)l,   // low 32 bits of a flat LDS addr == LDS offset
      0, 0);
#else
  *(float4*)l = *(const float4*)g;         // fallback: plain 128-bit copy
#endif
}

__device__ __forceinline__ void wait_async0() {
#if __has_builtin(__builtin_amdgcn_s_wait_asynccnt)
  __builtin_amdgcn_s_wait_asynccnt(0);
#elif __has_builtin(__builtin_amdgcn_global_load_async_to_lds_b128)
  asm volatile("s_wait_asynccnt 0" ::: "memory");
#endif
}

__global__ void __launch_bounds__(THREADS, 1)
bigabp_iter_kernel(const float* __restrict__ Hre, const float* __restrict__ Him,
                   const float* __restrict__ Xre, const float* __restrict__ Xim,
                   const float* __restrict__ Vx,  const float* __restrict__ Vh,
                   const float* __restrict__ Yre, const float* __restrict__ Yim,
                   const float* __restrict__ pN0, const float* __restrict__ pAl,
                   const float* __restrict__ pBe, const float* __restrict__ pGa,
                   const float* __restrict__ pEt, const float* __restrict__ pPm,
                   float* __restrict__ out, int nb)
{
  extern __shared__ float smem[];          // 6*ELEMS floats = 192 KB (dynamic LDS)

  const int t  = threadIdx.x;
  const int b  = blockIdx.x;
  const int nr = t >> 6;                   // [0,16)
  const int k  = t & 63;                   // [0,64)
  const size_t base = (size_t)b * ELEMS;

  // ---- stage the 6 big tensors into LDS with async b128 copies ----------
  const float* gsrc[6] = {Hre, Him, Xre, Xim, Vx, Vh};
#pragma unroll
  for (int j = 0; j < 6; ++j) {
#pragma unroll
    for (int r = 0; r < 2; ++r) {
      const int f = (r * THREADS + t) * 4;           // float index, 16B aligned
      async_g2l_b128(gsrc[j] + base + f, smem + j * ELEMS + f);
    }
  }

  // thread-local loads that never need LDS
  const float yr = Yre[(size_t)b * (NR_ * K_) + t];
  const float yi = Yim[(size_t)b * (NR_ * K_) + t];
  const float pm = pPm[(size_t)b * K_ + k];
  const float N0 = pN0[0], alpha = pAl[0], beta = pBe[0];
  const float gamma = pGa[0], eta = pEt[0];

  wait_async0();
  __syncthreads();

  // ---- pull this thread's (nr,k,:) slice into registers; NT sums local --
  float hr[NT_], hi[NT_], xr[NT_], xi[NT_], vX[NT_], vH[NT_];
  float hxr[NT_], hxi[NT_], tmp[NT_];
  float S_hxr = 0.f, S_hxi = 0.f, S_tmp = 0.f;
#pragma unroll
  for (int nt = 0; nt < NT_; ++nt) {
    const int e = nr * (NT_ * K_) + nt * K_ + k;
    hr[nt] = smem[0 * ELEMS + e]; hi[nt] = smem[1 * ELEMS + e];
    xr[nt] = smem[2 * ELEMS + e]; xi[nt] = smem[3 * ELEMS + e];
    vX[nt] = smem[4 * ELEMS + e]; vH[nt] = smem[5 * ELEMS + e];
    hxr[nt] = hr[nt] * xr[nt] - hi[nt] * xi[nt];
    hxi[nt] = hr[nt] * xi[nt] + hi[nt] * xr[nt];
    const float aH2 = hr[nt] * hr[nt] + hi[nt] * hi[nt];
    const float aX2 = xr[nt] * xr[nt] + xi[nt] * xi[nt];
    tmp[nt] = aH2 * vX[nt] + vH[nt] * (aX2 + vX[nt]);
    S_hxr += hxr[nt]; S_hxi += hxi[nt]; S_tmp += tmp[nt];
  }
  __syncthreads();   // inputs now in registers; re-alias LDS for reductions

  float* S_vt     = smem;          // [NT_*K_]  sums over nr
  float* S_te_re  = smem + 512;
  float* S_te_im  = smem + 1024;
  float* S_vth    = smem + 1536;   // [NR_*NT_] sums over k
  float* S_teh_re = smem + 1664;
  float* S_teh_im = smem + 1792;
  for (int i = t; i < 1920; i += THREADS) smem[i] = 0.f;
  __syncthreads();

  const float maskh = alpha * (1.f - pm) + beta * pm;

  // ---- accumulate cross-thread sums -------------------------------------
#pragma unroll
  for (int nt = 0; nt < NT_; ++nt) {
    const float er  = yr - S_hxr + hxr[nt];
    const float ei  = yi - S_hxi + hxi[nt];
    const float xiy = S_tmp - tmp[nt] + N0;
    const float rxx = 1.f / (xiy + vH[nt]);   // 1/xi_x
    const float rxh = 1.f / (xiy + vX[nt]);   // 1/xi_h (PHI=1)

    // X-message contributions: reduction over nr (cross-wave -> LDS atomics)
    const float vt    = (hr[nt] * hr[nt] + hi[nt] * hi[nt]) * rxx;
    const float te_re = (hr[nt] * er + hi[nt] * ei) * rxx;
    const float te_im = (hr[nt] * ei - hi[nt] * er) * rxx;
    atomicAdd(&S_vt   [nt * K_ + k], vt);
    atomicAdd(&S_te_re[nt * K_ + k], te_re);
    atomicAdd(&S_te_im[nt * K_ + k], te_im);

    // H-message contributions: reduction over k (wave32 butterfly + 2-way atomics)
    float vth    = maskh * (xr[nt] * xr[nt] + xi[nt] * xi[nt]) * rxh;
    float teh_re = maskh * (xr[nt] * er + xi[nt] * ei) * rxh;
    float teh_im = maskh * (xr[nt] * ei - xi[nt] * er) * rxh;
#pragma unroll
    for (int m = 16; m >= 1; m >>= 1) {
      vth    += __shfl_xor(vth,    m, 32);
      teh_re += __shfl_xor(teh_re, m, 32);
      teh_im += __shfl_xor(teh_im, m, 32);
    }
    if ((t & 31) == 0) {
      atomicAdd(&S_vth   [nr * NT_ + nt], vth);
      atomicAdd(&S_teh_re[nr * NT_ + nt], teh_re);
      atomicAdd(&S_teh_im[nr * NT_ + nt], teh_im);
    }
  }
  __syncthreads();

  // ---- leave-one-out, demod, blended updates, coalesced stores ----------
  const float s  = 0.70710678118654752440f;
  const float tg = (2.0f * s) / gamma;
  const size_t ps = (size_t)nb * ELEMS;    // output plane stride
  float* oHr = out + 0 * ps + base;
  float* oHi = out + 1 * ps + base;
  float* oXr = out + 2 * ps + base;
  float* oXi = out + 3 * ps + base;
  float* oVx = out + 4 * ps + base;
  float* oVh = out + 5 * ps + base;

#pragma unroll
  for (int nt = 0; nt < NT_; ++nt) {
    const int e = nr * (NT_ * K_) + nt * K_ + k;
    const float er  = yr - S_hxr + hxr[nt];
    const float ei  = yi - S_hxi + hxi[nt];
    const float xiy = S_tmp - tmp[nt] + N0;
    const float rxx = 1.f / (xiy + vH[nt]);
    const float rxh = 1.f / (xiy + vX[nt]);

    // --- VN_X ---
    const float vt    = (hr[nt] * hr[nt] + hi[nt] * hi[nt]) * rxx;
    const float te_re = (hr[nt] * er + hi[nt] * ei) * rxx;
    const float te_im = (hr[nt] * ei - hi[nt] * er) * rxx;
    const float varx = 1.f / (S_vt[nt * K_ + k] - vt);
    const float estr = (S_te_re[nt * K_ + k] - te_re) * varx;
    const float esti = (S_te_im[nt * K_ + k] - te_im) * varx;
    const float mr = s * tanhf(tg * estr);
    const float mi = s * tanhf(tg * esti);
    const float vpost = 1.f - (mr * mr + mi * mi);
    oXr[e] = xr[nt] + eta * ((mr    - xr[nt]) * pm);
    oXi[e] = xi[nt] + eta * ((mi    - xi[nt]) * pm);
    oVx[e] = vX[nt] + eta * ((vpost - vX[nt]) * pm);

    // --- VN_H ---
    const float vth    = maskh * (xr[nt] * xr[nt] + xi[nt] * xi[nt]) * rxh;
    const float teh_re = maskh * (xr[nt] * er + xi[nt] * ei) * rxh;
    const float teh_im = maskh * (xr[nt] * ei - xi[nt] * er) * rxh;
    const float varh = 1.f / (S_vth[nr * NT_ + nt] - vth);
    const float ehr  = (S_teh_re[nr * NT_ + nt] - teh_re) * varh;
    const float ehi  = (S_teh_im[nr * NT_ + nt] - teh_im) * varh;
    const float g    = 1.f / (varh + 1.f);    // PHI/(varh+PHI), PHI=1
    oHr[e] = hr[nt] + eta * (g * ehr  - hr[nt]);
    oHi[e] = hi[nt] + eta * (g * ehi  - hi[nt]);
    oVh[e] = vH[nt] + eta * (g * varh - vH[nt]);
  }
}

extern "C" void kernel_launch(void* const* d_in, const int* in_sizes, int n_in,
                              void* d_out, int out_size, void* d_ws, size_t ws_size,
                              hipStream_t stream) {
  (void)n_in; (void)out_size; (void)d_ws; (void)ws_size;
  const float* Hre = (const float*)d_in[0];
  const float* Him = (const float*)d_in[1];
  const float* Xre = (const float*)d_in[2];
  const float* Xim = (const float*)d_in[3];
  const float* Vx  = (const float*)d_in[4];
  const float* Vh  = (const float*)d_in[5];
  const float* Yre = (const float*)d_in[6];
  const float* Yim = (const float*)d_in[7];
  const float* N0  = (const float*)d_in[8];
  const float* al  = (const float*)d_in[9];
  const float* be  = (const float*)d_in[10];
  const float* ga  = (const float*)d_in[11];
  const float* et  = (const float*)d_in[12];
  const float* pmk = (const float*)d_in[13];

  const int nb = in_sizes[0] / ELEMS;          // B = 1024
  const size_t shmem = (size_t)6 * ELEMS * sizeof(float);  // 192 KB dynamic LDS

  hipLaunchKernelGGL(bigabp_iter_kernel, dim3(nb), dim3(THREADS), shmem, stream,
                     Hre, Him, Xre, Xim, Vx, Vh, Yre, Yim,
                     N0, al, be, ga, et, pmk, (float*)d_out, nb);
}